// NodeGraphSAGE_13984413516530
// MI455X (gfx1250) — compile-verified
//
#include <hip/hip_runtime.h>
#include <math.h>

// ---------------------------------------------------------------------------
// 2-layer GraphSAGE (mean aggregation), N=100000 nodes, E=625000 edges, C=128.
//   h   = sigmoid(mean1 @ W1l^T + b1l + x @ W1r^T)
//   out =         mean2 @ W2l^T + b2l + h @ W2r^T
// Scatter phases: fp32 global atomics (resolve in 192MB L2).
// GEMM phases:    V_WMMA_F32_16X16X4_F32 (full fp32 fidelity).
// ---------------------------------------------------------------------------

typedef __attribute__((ext_vector_type(2))) float v2f;
typedef __attribute__((ext_vector_type(8))) float v8f;

#define CCH 128           // channels (in = hid = out = 128)

// ----------------------------- degree count --------------------------------
__global__ __launch_bounds__(256) void degree_kernel(
    const int* __restrict__ dst, float* __restrict__ cnt, int E) {
  int e = blockIdx.x * blockDim.x + threadIdx.x;
  if (e < E) {
    unsafeAtomicAdd(&cnt[dst[e]], 1.0f);
  }
}

__global__ __launch_bounds__(256) void inv_kernel(
    const float* __restrict__ cnt, float* __restrict__ inv, int n) {
  int i = blockIdx.x * blockDim.x + threadIdx.x;
  if (i < n) inv[i] = 1.0f / fmaxf(cnt[i], 1.0f);
}

// --------------------------- edge scatter-add ------------------------------
// One thread handles 4 channels of one edge: float4 gather + 4 fp32 atomics.
__global__ __launch_bounds__(256) void scatter_kernel(
    const float* __restrict__ feat, const int* __restrict__ src,
    const int* __restrict__ dst, float* __restrict__ agg, int E) {
  long tid = (long)blockIdx.x * blockDim.x + threadIdx.x;
  long total = (long)E * 32;
  if (tid >= total) return;
  int e = (int)(tid >> 5);
  int c = ((int)tid & 31) * 4;
  int s = src[e];
  int d = dst[e];
  const float4 v = *(const float4*)(feat + (long)s * CCH + c);
  float* base = agg + (long)d * CCH + c;
  unsafeAtomicAdd(base + 0, v.x);
  unsafeAtomicAdd(base + 1, v.y);
  unsafeAtomicAdd(base + 2, v.z);
  unsafeAtomicAdd(base + 3, v.w);
}

// ------------------------- fused SAGE layer (WMMA) -------------------------
// out[m][:] = (agg[m]*inv[m]) @ Wl^T + bl + xin[m] @ Wr^T   (+ sigmoid)
// Block = 128 threads = 4 waves; each wave owns 32 nodes (2 M-tiles) and all
// 8 N-tiles (128 outputs) -> 16 fp32 accumulator fragments per wave.
template <bool SIG>
__global__ __launch_bounds__(128) void sage_layer_kernel(
    const float* __restrict__ xin,   // [N][128] root features
    const float* __restrict__ agg,   // [N][128] neighbor sums
    const float* __restrict__ inv,   // [N]      1/max(deg,1)
    const float* __restrict__ Wl,    // [128][128] row-major [out][in]
    const float* __restrict__ bl,    // [128]
    const float* __restrict__ Wr,    // [128][128]
    float* __restrict__ out,         // [N][128]
    int n_nodes) {
  const int lane = threadIdx.x & 31;
  const int wave = threadIdx.x >> 5;
  const int kk   = lane >> 4;   // 0/1 : which K-pair of the 16x4 fragment
  const int ml   = lane & 15;   // row (A) / col (B,C,D) within tile
  const int mbase = blockIdx.x * 128 + wave * 32;

  // Clamp node ids so OOB waves still issue valid loads (EXEC must be all-1s
  // around WMMA); stores are predicated instead.
  int m0 = mbase + ml;       if (m0 >= n_nodes) m0 = n_nodes - 1;
  int m1 = mbase + 16 + ml;  if (m1 >= n_nodes) m1 = n_nodes - 1;
  const float inv0 = inv[m0];
  const float inv1 = inv[m1];

  const float2* a0p = (const float2*)(agg + (long)m0 * CCH);
  const float2* a1p = (const float2*)(agg + (long)m1 * CCH);
  const float2* x0p = (const float2*)(xin + (long)m0 * CCH);
  const float2* x1p = (const float2*)(xin + (long)m1 * CCH);

  v8f acc[2][8];
#pragma unroll
  for (int t = 0; t < 2; ++t)
#pragma unroll
    for (int n = 0; n < 8; ++n) acc[t][n] = {};

  // ---- first half of K: mean (= agg * inv) against Wl ----
  for (int ks = 0; ks < 32; ++ks) {
    const int kh = ks * 2 + kk;          // float2 index; covers k = 2*kh, 2*kh+1
    float2 r0 = a0p[kh];
    float2 r1 = a1p[kh];
    v2f A0; A0.x = r0.x * inv0; A0.y = r0.y * inv0;
    v2f A1; A1.x = r1.x * inv1; A1.y = r1.y * inv1;
#pragma unroll
    for (int n = 0; n < 8; ++n) {
      float2 bw = *(const float2*)(Wl + (long)(n * 16 + ml) * CCH + 2 * kh);
      v2f B; B.x = bw.x; B.y = bw.y;
      acc[0][n] = __builtin_amdgcn_wmma_f32_16x16x4_f32(
          false, A0, false, B, (short)0, acc[0][n], false, false);
      acc[1][n] = __builtin_amdgcn_wmma_f32_16x16x4_f32(
          false, A1, false, B, (short)0, acc[1][n], false, false);
    }
  }

  // ---- second half of K: root features against Wr ----
  for (int ks = 0; ks < 32; ++ks) {
    const int kh = ks * 2 + kk;
    float2 r0 = x0p[kh];
    float2 r1 = x1p[kh];
    v2f A0; A0.x = r0.x; A0.y = r0.y;
    v2f A1; A1.x = r1.x; A1.y = r1.y;
#pragma unroll
    for (int n = 0; n < 8; ++n) {
      float2 bw = *(const float2*)(Wr + (long)(n * 16 + ml) * CCH + 2 * kh);
      v2f B; B.x = bw.x; B.y = bw.y;
      acc[0][n] = __builtin_amdgcn_wmma_f32_16x16x4_f32(
          false, A0, false, B, (short)0, acc[0][n], false, false);
      acc[1][n] = __builtin_amdgcn_wmma_f32_16x16x4_f32(
          false, A1, false, B, (short)0, acc[1][n], false, false);
    }
  }

  // ---- epilogue: bias (+ sigmoid), predicated stores ----
#pragma unroll
  for (int n = 0; n < 8; ++n) {
    const float bias = bl[n * 16 + ml];
#pragma unroll
    for (int t = 0; t < 2; ++t) {
#pragma unroll
      for (int r = 0; r < 8; ++r) {
        int m = mbase + t * 16 + r + 8 * kk;   // C/D row for VGPR r
        if (m < n_nodes) {
          float v = acc[t][n][r] + bias;
          if (SIG) v = 1.0f / (1.0f + __expf(-v));
          out[(long)m * CCH + n * 16 + ml] = v;
        }
      }
    }
  }
}

// ---------------------------------------------------------------------------
extern "C" void kernel_launch(void* const* d_in, const int* in_sizes, int n_in,
                              void* d_out, int out_size, void* d_ws, size_t ws_size,
                              hipStream_t stream) {
  const float* x   = (const float*)d_in[0];
  const int*   ei  = (const int*)d_in[1];   // [2][E], int32 (JAX default x64 off)
  const float* W1l = (const float*)d_in[2];
  const float* b1l = (const float*)d_in[3];
  const float* W1r = (const float*)d_in[4];
  const float* W2l = (const float*)d_in[5];
  const float* b2l = (const float*)d_in[6];
  const float* W2r = (const float*)d_in[7];
  float* out = (float*)d_out;

  const int N = in_sizes[0] / CCH;   // 100000
  const int E = in_sizes[1] / 2;     // 625000

  // Workspace layout
  float* agg = (float*)d_ws;                       // N*128
  float* h   = agg + (size_t)N * CCH;              // N*128
  float* cnt = h + (size_t)N * CCH;                // N
  float* vin = cnt + (size_t)N;                    // N

  const int* src = ei;
  const int* dst = ei + E;

  // ---- shared degree info (same graph both layers) ----
  hipMemsetAsync(agg, 0, (size_t)N * CCH * sizeof(float), stream);
  hipMemsetAsync(cnt, 0, (size_t)N * sizeof(float), stream);
  degree_kernel<<<(E + 255) / 256, 256, 0, stream>>>(dst, cnt, E);
  inv_kernel<<<(N + 255) / 256, 256, 0, stream>>>(cnt, vin, N);

  const long sthreads = (long)E * 32;
  const int  sblocks  = (int)((sthreads + 255) / 256);
  const int  lblocks  = (N + 127) / 128;

  // ---- layer 1 ----
  scatter_kernel<<<sblocks, 256, 0, stream>>>(x, src, dst, agg, E);
  sage_layer_kernel<true><<<lblocks, 128, 0, stream>>>(
      x, agg, vin, W1l, b1l, W1r, h, N);

  // ---- layer 2 ----
  hipMemsetAsync(agg, 0, (size_t)N * CCH * sizeof(float), stream);
  scatter_kernel<<<sblocks, 256, 0, stream>>>(h, src, dst, agg, E);
  sage_layer_kernel<false><<<lblocks, 128, 0, stream>>>(
      h, agg, vin, W2l, b2l, W2r, out, N);
}